// VolumeRaycaster_72756745994794
// MI455X (gfx1250) — compile-verified
//
#include <hip/hip_runtime.h>

// Volume raycaster for MI455X (gfx1250, wave32).
// vol: (4, 4, 128,128,128) f32  [ch 0..2 = color, ch 3 = density]
// tfm: (4, 4, 4) f32, pure Y-rotation
// out: (4, 3, 240, 320) f32
//
// Gather-bound: the whole dataset fits in the 192MB L2 -> L2-bandwidth /
// VMEM-issue bound. Plan:
//   pass 1 (repack): planar 4x8MB channel planes -> AoS float4 voxels in d_ws
//                    (one-time ~256MB HBM traffic ~= 11us @ 23.3 TB/s),
//                    planar reads marked non-temporal to protect L2.
//   pass 2 (raycast): 1 thread/ray, 8x global_load_b128 per trilinear sample
//                     (vs 16x b64 planar), outside-volume skipping, early ray
//                     termination, wave = 32 adjacent pixels in a row for
//                     cache-line coherent gathers.
// Fallback planar kernel if ws_size < 128MB.

#define WRES 320
#define HRES 240
#define NSAMP 256
#define VD 128
#define ZS (VD * VD)          // 16384
#define V3 (VD * VD * VD)     // 2097152 = 2^21
#define HW (WRES * HRES)      // 76800
#define DENS_K (100.0f / 256.0f)

// ---------------------------------------------------------------------------
// Pass 1: repack planar channels -> float4 AoS (x,y,z,w) = (r,g,b,density)
// ---------------------------------------------------------------------------
__global__ __launch_bounds__(256)
void repack_kernel(const float* __restrict__ vol, float4* __restrict__ pk) {
    int t = blockIdx.x * 256 + threadIdx.x;      // [0, 4*V3)
    int b = t >> 21;                             // V3 == 2^21
    int r = t & (V3 - 1);
    const float* vb = vol + (size_t)b * 4 * V3;
    float4 v;
    v.x = __builtin_nontemporal_load(vb + 0 * V3 + r);
    v.y = __builtin_nontemporal_load(vb + 1 * V3 + r);
    v.z = __builtin_nontemporal_load(vb + 2 * V3 + r);
    v.w = __builtin_nontemporal_load(vb + 3 * V3 + r);
    pk[t] = v;                                   // coalesced b128 stores
}

// ---------------------------------------------------------------------------
// Packed helpers
// ---------------------------------------------------------------------------
__device__ __forceinline__ float4 lerp4(float4 a, float4 b, float t) {
    float4 r;
    r.x = fmaf(t, b.x - a.x, a.x);
    r.y = fmaf(t, b.y - a.y, a.y);
    r.z = fmaf(t, b.z - a.z, a.z);
    r.w = fmaf(t, b.w - a.w, a.w);
    return r;
}

__device__ __forceinline__ float4 tri8p(const float4* __restrict__ p, int base,
                                        float fx, float fy, float fz) {
    float4 a0 = p[base],           a1 = p[base + 1];
    float4 b0 = p[base + VD],      b1 = p[base + VD + 1];
    float4 c0 = p[base + ZS],      c1 = p[base + ZS + 1];
    float4 d0 = p[base + ZS + VD], d1 = p[base + ZS + VD + 1];
    float4 xa = lerp4(a0, a1, fx);
    float4 xb = lerp4(b0, b1, fx);
    float4 xc = lerp4(c0, c1, fx);
    float4 xd = lerp4(d0, d1, fx);
    float4 ya = lerp4(xa, xb, fy);
    float4 yb = lerp4(xc, xd, fy);
    return lerp4(ya, yb, fz);
}

// Boundary path: exact reference clamp+validity semantics, <=8 b128 loads.
__device__ __forceinline__ float4 sample_masked_p(const float4* __restrict__ p,
                                                  float vx, float vy, float vz) {
    float fxf = floorf(vx), fyf = floorf(vy), fzf = floorf(vz);
    float fx = vx - fxf, fy = vy - fyf, fz = vz - fzf;
    int xi0 = (int)fxf, yi0 = (int)fyf, zi0 = (int)fzf;
    float wx[2] = {1.0f - fx, fx};
    float wy[2] = {1.0f - fy, fy};
    float wz[2] = {1.0f - fz, fz};
    float4 acc = make_float4(0.0f, 0.0f, 0.0f, 0.0f);
    #pragma unroll
    for (int dz = 0; dz < 2; ++dz) {
        int z = zi0 + dz;
        if ((unsigned)z >= VD) continue;
        #pragma unroll
        for (int dy = 0; dy < 2; ++dy) {
            int y = yi0 + dy;
            if ((unsigned)y >= VD) continue;
            #pragma unroll
            for (int dx = 0; dx < 2; ++dx) {
                int x = xi0 + dx;
                if ((unsigned)x >= VD) continue;
                float w = wx[dx] * wy[dy] * wz[dz];
                float4 v = p[z * ZS + y * VD + x];
                acc.x = fmaf(w, v.x, acc.x);
                acc.y = fmaf(w, v.y, acc.y);
                acc.z = fmaf(w, v.z, acc.z);
                acc.w = fmaf(w, v.w, acc.w);
            }
        }
    }
    return acc;
}

// ---------------------------------------------------------------------------
// Pass 2: raycast over packed volume
// ---------------------------------------------------------------------------
__global__ __launch_bounds__(256)
void raycast_packed(const float4* __restrict__ pk, const float* __restrict__ tfm,
                    float* __restrict__ out) {
    const int blocksPerBatch = HW / 256;         // 300 -> b uniform per block
    int b   = blockIdx.x / blocksPerBatch;
    int rem = blockIdx.x - b * blocksPerBatch;
    int pix = rem * 256 + threadIdx.x;
    int px  = pix % WRES;
    int py  = pix / WRES;

    float cA = tfm[b * 16 + 0];                  // row0 = [c,0,s,0]
    float sA = tfm[b * 16 + 2];                  // row2 = [-s,0,c,0]
    const float4* __restrict__ vb = pk + (size_t)b * V3;
    float* __restrict__ outp = out + (size_t)b * 3 * HW + pix;

    float xn = fmaf((float)px, 2.0f / (WRES - 1), -1.0f);
    float yn = fmaf((float)py, 2.0f / (HRES - 1), -1.0f);

    // y invariant along ray (pure Y-rotation).
    float vy  = fmaf(yn, 64.0f, 63.5f);
    float fyf = floorf(vy);
    int   y0  = (int)fyf;
    float fy  = vy - fyf;
    bool fastY = ((unsigned)y0 <= VD - 2);
    int  yrow  = y0 * VD;

    // vx(zn) = 64*sA*zn + (64*cA*xn + 63.5); vz(zn) = 64*cA*zn + (-64*sA*xn + 63.5)
    float sA64 = sA * 64.0f, cA64 = cA * 64.0f;
    float vxc = fmaf(cA * xn, 64.0f, 63.5f);
    float vzc = fmaf(-sA * xn, 64.0f, 63.5f);

    float T = 1.0f;
    float accR = 0.0f, accG = 0.0f, accB = 0.0f, accW = 0.0f;

    #pragma unroll 2
    for (int s = 0; s < NSAMP; ++s) {
        float zn = fmaf((float)s, 2.0f / 255.0f, -1.0f);
        float vx = fmaf(zn, sA64, vxc);
        float vz = fmaf(zn, cA64, vzc);
        float fxf = floorf(vx), fzf = floorf(vz);
        int x0 = (int)fxf, z0 = (int)fzf;

        // All 8 corners invalid on x or z -> ds == 0 -> exact skip.
        if ((unsigned)(x0 + 1) > VD || (unsigned)(z0 + 1) > VD) continue;

        float4 smp;
        if (fastY && (unsigned)x0 <= VD - 2 && (unsigned)z0 <= VD - 2) {
            float fx = vx - fxf, fz = vz - fzf;
            int base = z0 * ZS + yrow + x0;
            smp = tri8p(vb, base, fx, fy, fz);   // 8x global_load_b128
        } else {
            smp = sample_masked_p(vb, vx, vy, vz);
        }

        float ds = smp.w * DENS_K;
        T *= (1.0f - ds);                        // transmission incl. current
        float w = ds * T;
        accR = fmaf(w, smp.x, accR);
        accG = fmaf(w, smp.y, accG);
        accB = fmaf(w, smp.z, accB);
        accW += w;

        // Remaining weights ~ T < 1e-7: below fp32 noise. Deterministic.
        if (T < 1e-7f) break;
    }

    float scale = (1.0f - T) / (accW + 1e-6f);
    outp[0]      = accR * scale;
    outp[HW]     = accG * scale;
    outp[2 * HW] = accB * scale;
}

// ---------------------------------------------------------------------------
// Fallback: planar 16x b64 version (used only if d_ws < 128MB)
// ---------------------------------------------------------------------------
struct __attribute__((packed, aligned(4))) F2 { float lo, hi; };

__device__ __forceinline__ float tri8(const float* __restrict__ p, int base,
                                      float fx, float fy, float fz) {
    F2 a = *(const F2*)(p + base);
    F2 b = *(const F2*)(p + base + VD);
    F2 c = *(const F2*)(p + base + ZS);
    F2 d = *(const F2*)(p + base + ZS + VD);
    float x00 = fmaf(fx, a.hi - a.lo, a.lo);
    float x01 = fmaf(fx, b.hi - b.lo, b.lo);
    float x10 = fmaf(fx, c.hi - c.lo, c.lo);
    float x11 = fmaf(fx, d.hi - d.lo, d.lo);
    float y0v = fmaf(fy, x01 - x00, x00);
    float y1v = fmaf(fy, x11 - x10, x10);
    return fmaf(fz, y1v - y0v, y0v);
}

__device__ __noinline__ void sample_masked(const float* __restrict__ vb,
                                           float vx, float vy, float vz,
                                           float& r, float& g, float& bl, float& dn) {
    float fxf = floorf(vx), fyf = floorf(vy), fzf = floorf(vz);
    float fx = vx - fxf, fy = vy - fyf, fz = vz - fzf;
    int xi0 = (int)fxf, yi0 = (int)fyf, zi0 = (int)fzf;
    float wx[2] = {1.0f - fx, fx};
    float wy[2] = {1.0f - fy, fy};
    float wz[2] = {1.0f - fz, fz};
    r = 0.0f; g = 0.0f; bl = 0.0f; dn = 0.0f;
    #pragma unroll
    for (int dz = 0; dz < 2; ++dz) {
        int z = zi0 + dz;
        if ((unsigned)z >= VD) continue;
        #pragma unroll
        for (int dy = 0; dy < 2; ++dy) {
            int y = yi0 + dy;
            if ((unsigned)y >= VD) continue;
            #pragma unroll
            for (int dx = 0; dx < 2; ++dx) {
                int x = xi0 + dx;
                if ((unsigned)x >= VD) continue;
                float w = wx[dx] * wy[dy] * wz[dz];
                int idx = z * ZS + y * VD + x;
                dn = fmaf(w, vb[3 * V3 + idx], dn);
                r  = fmaf(w, vb[0 * V3 + idx], r);
                g  = fmaf(w, vb[1 * V3 + idx], g);
                bl = fmaf(w, vb[2 * V3 + idx], bl);
            }
        }
    }
}

__global__ __launch_bounds__(256)
void raycast_planar(const float* __restrict__ vol, const float* __restrict__ tfm,
                    float* __restrict__ out) {
    const int blocksPerBatch = HW / 256;
    int b   = blockIdx.x / blocksPerBatch;
    int rem = blockIdx.x - b * blocksPerBatch;
    int pix = rem * 256 + threadIdx.x;
    int px  = pix % WRES;
    int py  = pix / WRES;

    float cA = tfm[b * 16 + 0];
    float sA = tfm[b * 16 + 2];
    const float* __restrict__ vb = vol + (size_t)b * 4 * V3;
    float* __restrict__ outp = out + (size_t)b * 3 * HW + pix;

    float xn = fmaf((float)px, 2.0f / (WRES - 1), -1.0f);
    float yn = fmaf((float)py, 2.0f / (HRES - 1), -1.0f);

    float vy  = fmaf(yn, 64.0f, 63.5f);
    float fyf = floorf(vy);
    int   y0  = (int)fyf;
    float fy  = vy - fyf;
    bool fastY = ((unsigned)y0 <= VD - 2);
    int  yrow  = y0 * VD;

    float sA64 = sA * 64.0f, cA64 = cA * 64.0f;
    float vxc = fmaf(cA * xn, 64.0f, 63.5f);
    float vzc = fmaf(-sA * xn, 64.0f, 63.5f);

    float T = 1.0f;
    float accR = 0.0f, accG = 0.0f, accB = 0.0f, accW = 0.0f;

    #pragma unroll 2
    for (int s = 0; s < NSAMP; ++s) {
        float zn = fmaf((float)s, 2.0f / 255.0f, -1.0f);
        float vx = fmaf(zn, sA64, vxc);
        float vz = fmaf(zn, cA64, vzc);
        float fxf = floorf(vx), fzf = floorf(vz);
        int x0 = (int)fxf, z0 = (int)fzf;

        if ((unsigned)(x0 + 1) > VD || (unsigned)(z0 + 1) > VD) continue;

        float r, g, bl, dn;
        if (fastY && (unsigned)x0 <= VD - 2 && (unsigned)z0 <= VD - 2) {
            float fx = vx - fxf, fz = vz - fzf;
            int base = z0 * ZS + yrow + x0;
            dn = tri8(vb + 3 * V3, base, fx, fy, fz);
            r  = tri8(vb + 0 * V3, base, fx, fy, fz);
            g  = tri8(vb + 1 * V3, base, fx, fy, fz);
            bl = tri8(vb + 2 * V3, base, fx, fy, fz);
        } else {
            sample_masked(vb, vx, vy, vz, r, g, bl, dn);
        }

        float ds = dn * DENS_K;
        T *= (1.0f - ds);
        float w = ds * T;
        accR = fmaf(w, r, accR);
        accG = fmaf(w, g, accG);
        accB = fmaf(w, bl, accB);
        accW += w;
        if (T < 1e-7f) break;
    }

    float scale = (1.0f - T) / (accW + 1e-6f);
    outp[0]      = accR * scale;
    outp[HW]     = accG * scale;
    outp[2 * HW] = accB * scale;
}

// ---------------------------------------------------------------------------
extern "C" void kernel_launch(void* const* d_in, const int* in_sizes, int n_in,
                              void* d_out, int out_size, void* d_ws, size_t ws_size,
                              hipStream_t stream) {
    (void)in_sizes; (void)n_in; (void)out_size;
    const float* vol = (const float*)d_in[0];
    const float* tfm = (const float*)d_in[1];
    float* out = (float*)d_out;

    const int blocksPerBatch = HW / 256;         // 300
    const int rayBlocks = 4 * blocksPerBatch;    // 1200 blocks x 8 wave32
    const size_t packedBytes = (size_t)4 * V3 * sizeof(float4);  // 128 MB

    if (d_ws != nullptr && ws_size >= packedBytes) {
        float4* pk = (float4*)d_ws;
        repack_kernel<<<(4 * V3) / 256, 256, 0, stream>>>(vol, pk);
        raycast_packed<<<rayBlocks, 256, 0, stream>>>(pk, tfm, out);
    } else {
        raycast_planar<<<rayBlocks, 256, 0, stream>>>(vol, tfm, out);
    }
}